// OpponentModelOracle_18829136626176
// MI455X (gfx1250) — compile-verified
//
#include <hip/hip_runtime.h>
#include <stdint.h>

// ---------------------------------------------------------------------------
// OpponentModelOracle for MI455X (gfx1250), compile-only loop.
// Memory-bound (~320 MB traffic -> ~14us roofline @ 23.3 TB/s). No matmul in
// the reference, so no WMMA; the CDNA5 path used is the Tensor Data Mover
// (tensor_load_to_lds + s_wait_tensorcnt) to stream the two needed one-hot
// channels into LDS, plus non-temporal float4 output stores.
// ---------------------------------------------------------------------------

typedef unsigned int v4u __attribute__((ext_vector_type(4)));
typedef int          v4i __attribute__((ext_vector_type(4)));
typedef int          v8i __attribute__((ext_vector_type(8)));
typedef float        v4f __attribute__((ext_vector_type(4)));

__device__ __forceinline__ unsigned umin2(unsigned a, unsigned b) { return a < b ? a : b; }
__device__ __forceinline__ unsigned umax2(unsigned a, unsigned b) { return a > b ? a : b; }

struct Smem {
  float    food[4096];   // channel 1 of each cell
  float    opp[4096];    // channel 3 of each cell
  unsigned rA[256];      // food-count reduction
  unsigned rB[256];      // min-opp-index reduction
  unsigned rC[256];      // smallest key
  unsigned rD[256];      // 2nd smallest key
};                        // 36,864 bytes

#if __has_builtin(__builtin_amdgcn_tensor_load_to_lds)
#define ORACLE_HAS_TDM 1
#endif

#ifdef ORACLE_HAS_TDM
// DMA a strided channel: 4096 rows x 1 f32 element, row stride 16 B, into
// a contiguous 16 KB LDS region. Descriptor bitfields per CDNA5 ISA ch.8.
__device__ __forceinline__ void tdm_load_channel(unsigned long long gaddr, unsigned lds_off) {
  v4u g0;
  g0.x = 1u;                                                  // count=1, user mode
  g0.y = lds_off;                                             // lds_addr (bytes)
  g0.z = (unsigned)(gaddr & 0xFFFFFFFFull);                   // global_addr[31:0]
  g0.w = (unsigned)((gaddr >> 32) & 0x01FFFFFFull)            // global_addr[56:32]
         | 0x80000000u;                                       // type=2 ("image")
  v8i g1;
  g1[0] = (int)(2u << 16);      // data_size=2 -> 4-byte elements; mask=0
  g1[1] = (int)(1u << 16);      // tensor_dim0 = 1 (bits 63:48)
  g1[2] = (int)(4096u << 16);   // tensor_dim1 = 4096 (bits 95:80)
  g1[3] = (int)(1u << 16);      // tile_dim0 = 1 (bits 127:112)
  g1[4] = (int)4096u;           // tile_dim1 = 4096, tile_dim2 = 0
  g1[5] = (int)4u;              // tensor_dim0_stride = 4 elements = 16 bytes
  g1[6] = 0;                    // stride0 hi / stride1 lo
  g1[7] = 0;                    // stride1 hi
  v4i gz = {0, 0, 0, 0};
#if defined(__clang_major__) && (__clang_major__ >= 23)
  v8i gz8 = {0, 0, 0, 0, 0, 0, 0, 0};
  __builtin_amdgcn_tensor_load_to_lds(g0, g1, gz, gz, gz8, 0);
#else
  __builtin_amdgcn_tensor_load_to_lds(g0, g1, gz, gz, 0);
#endif
}
#endif

__global__ __launch_bounds__(256) void oracle_kernel(const float* __restrict__ x,
                                                     float* __restrict__ out) {
  __shared__ Smem sm;
  const unsigned tid = threadIdx.x;
  const unsigned b   = blockIdx.x;

  // ---- Stage the food/opp channels of this batch's 64x64 grid into LDS ----
#ifdef ORACLE_HAS_TDM
  if (tid < 32) {  // one wave issues the DMA; EXEC all-ones inside this branch
    unsigned long long base =
        (unsigned long long)(uintptr_t)x + (unsigned long long)b * 65536ull;
    // Generic pointers to LDS carry the byte offset in their low 32 bits.
    unsigned food_off = (unsigned)(uintptr_t)&sm.food[0];
    unsigned opp_off  = (unsigned)(uintptr_t)&sm.opp[0];
    tdm_load_channel(base + 4ull,  food_off);   // channel 1 (Food)
    tdm_load_channel(base + 12ull, opp_off);    // channel 3 (Opponent)
    __builtin_amdgcn_s_wait_tensorcnt(0);
  }
#else
  {
    const v4f* cells = (const v4f*)x;
    for (int c = 0; c < 16; ++c) {
      unsigned idx = (unsigned)c * 256u + tid;
      v4f cell     = cells[(size_t)b * 4096u + idx];
      sm.food[idx] = cell.y;
      sm.opp[idx]  = cell.w;
    }
  }
#endif
  __syncthreads();

  // ---- Phase A: per-thread food bits, food count, min opponent index ----
  const v4f* foodv = (const v4f*)sm.food;
  const v4f* oppv  = (const v4f*)sm.opp;

  unsigned foodbits = 0u, oppmin = 0xFFFFFFFFu, cnt = 0u;
#pragma unroll
  for (int g = 0; g < 4; ++g) {
    v4f f = foodv[g * 256 + tid];
    v4f o = oppv[g * 256 + tid];
    unsigned baseidx = (unsigned)g * 1024u + tid * 4u;
#pragma unroll
    for (int e = 0; e < 4; ++e) {
      if (f[e] == 1.0f) { foodbits |= 1u << (g * 4 + e); cnt++; }
      if (o[e] == 1.0f) oppmin = umin2(oppmin, baseidx + (unsigned)e);
    }
  }

  sm.rA[tid] = cnt;
  sm.rB[tid] = oppmin;
  __syncthreads();
#pragma unroll
  for (int s = 128; s > 0; s >>= 1) {
    if (tid < (unsigned)s) {
      sm.rA[tid] += sm.rA[tid + s];
      sm.rB[tid]  = umin2(sm.rB[tid], sm.rB[tid + s]);
    }
    __syncthreads();
  }
  const unsigned n_food = sm.rA[0];
  const unsigned oppidx = sm.rB[0];

  const bool has_opp = (oppidx != 0xFFFFFFFFu);
  const int  oi = has_opp ? (int)(oppidx >> 6) : 0;   // matches argmax()==0 fallback
  const int  oj = has_opp ? (int)(oppidx & 63u) : 0;

  // ---- Phase B: two smallest (d^2, idx) keys over food cells -------------
  // d^2 <= 63^2+63^2 = 7938 < 2^13 and idx < 2^12, so key = (d2<<12)|idx is
  // an exact lexicographic (distance, row-major index) order. sqrtf at the
  // end reproduces the reference's f32 sqrt bit-for-bit (d2 exact in f32).
  unsigned k0 = 0xFFFFFFFFu, k1 = 0xFFFFFFFFu;
#pragma unroll
  for (int g = 0; g < 4; ++g) {
#pragma unroll
    for (int e = 0; e < 4; ++e) {
      if ((foodbits >> (g * 4 + e)) & 1u) {
        unsigned idx = (unsigned)g * 1024u + tid * 4u + (unsigned)e;
        int di = (int)(idx >> 6) - oi;
        int dj = (int)(idx & 63u) - oj;
        unsigned d2  = (unsigned)(di * di + dj * dj);
        unsigned key = (d2 << 12) | idx;
        if (key < k0) { k1 = k0; k0 = key; }
        else          { k1 = umin2(k1, key); }
      }
    }
  }
  sm.rC[tid] = k0;
  sm.rD[tid] = k1;
  __syncthreads();
#pragma unroll
  for (int s = 128; s > 0; s >>= 1) {
    if (tid < (unsigned)s) {
      unsigned a0 = sm.rC[tid],     a1 = sm.rD[tid];
      unsigned b0 = sm.rC[tid + s], b1 = sm.rD[tid + s];
      sm.rC[tid] = umin2(a0, b0);
      sm.rD[tid] = umin2(umin2(a1, b1), umax2(a0, b0));
    }
    __syncthreads();
  }
  const unsigned K0 = sm.rC[0];
  const unsigned K1 = sm.rD[0];

  // ---- Phase C: flags + streamed output ----------------------------------
  const unsigned nearest = K0 & 0xFFFu;
  const float d0   = sqrtf((float)(K0 >> 12));
  const float d1   = sqrtf((float)(K1 >> 12));
  const float diff = d1 - d0;

  const bool many      = (n_food > 1u);
  const bool notstart  = !(oi == 3 && oj == 6);
  const bool cond_a    = many && has_opp && notstart;
  const bool ambiguous = many && (cond_a ? (diff < 0.1f) : true);
  const bool single    = (many && cond_a && (diff >= 0.1f)) || (n_food == 1u);

  v4f* outb = (v4f*)(out + (size_t)b * 4096u);
#pragma unroll
  for (int g = 0; g < 4; ++g) {
    unsigned baseidx = (unsigned)g * 1024u + tid * 4u;
    v4f pv;
#pragma unroll
    for (int e = 0; e < 4; ++e) {
      bool set;
      if (ambiguous)   set = (foodbits >> (g * 4 + e)) & 1u;
      else if (single) set = (baseidx + (unsigned)e == nearest);
      else             set = false;
      pv[e] = set ? 10.0f : -10.0f;
    }
    __builtin_nontemporal_store(pv, &outb[g * 256 + tid]);
  }
}

extern "C" void kernel_launch(void* const* d_in, const int* in_sizes, int n_in,
                              void* d_out, int out_size, void* d_ws, size_t ws_size,
                              hipStream_t stream) {
  (void)n_in; (void)out_size; (void)d_ws; (void)ws_size;
  const float* x = (const float*)d_in[0];
  float* out = (float*)d_out;
  const int B = in_sizes[0] / (64 * 64 * 4);   // 4096
  hipLaunchKernelGGL(oracle_kernel, dim3((unsigned)B), dim3(256), 0, stream, x, out);
}